// gtnet_11527692223292
// MI455X (gfx1250) — compile-verified
//
#include <hip/hip_runtime.h>

typedef __attribute__((ext_vector_type(2))) float v2f;
typedef __attribute__((ext_vector_type(4))) float v4f;
typedef __attribute__((ext_vector_type(8))) float v8f;

namespace {
constexpr int cB = 32, cN = 1000, cBN = cB * cN;       // batch, nodes, columns
constexpr int cSEQ = 12, cRF = 19, cIN = 2;
constexpr int cRC = 32, cSC = 64, cEC = 128, cOUT = 12;
constexpr int cND = 40, cK = 20;
constexpr float cALPHA = 3.0f, cPROP = 0.05f;
}

// ---------------------------------------------------------------------------
// f32 WMMA GEMM, templated on MT (= M/16 tiles, fully resident accumulators)
// and D (compile-time row-mapping divisor).
//   Y[M][cols] (=|+=|relu=) W[M][K] * X + bias
//   X element offset(r) = (r/D)*strideC + (r%D + t0)*ncols + col
// One B fragment feeds MT WMMAs. A operand is a contiguous b64 load.
// blockDim (32,4): 4 column tiles per block. No EXEC masking (M % 16 == 0).
// ---------------------------------------------------------------------------
template <int MT, int D>
__global__ void wmma_gemm_f32(const float* __restrict__ Wm, const float* __restrict__ X,
                              float* __restrict__ Y, const float* __restrict__ bias,
                              int K, int strideC, int t0, int ncols, int ldY, int mode) {
  const int n0 = (blockIdx.x * blockDim.y + threadIdx.y) * 16;
  const int lane = threadIdx.x;
  const int lm = lane & 15;
  const int hi = lane >> 4;
  v8f acc[MT];
#pragma unroll
  for (int m = 0; m < MT; ++m) acc[m] = (v8f){};
  const float* Wp = Wm + (size_t)lm * K;                     // A row base (tile 0)
  const float* Xb = X + (size_t)t0 * ncols + n0 + lm;        // column base
  for (int kb = 0; kb < K; kb += 4) {
    const int ka = kb + hi * 2;            // lanes 0-15: rows k,k+1 ; 16-31: k+2,k+3
    v2f b;
    {
      const int qx = ka / D, rx = ka % D;               // compile-time D -> cheap
      const int qy = (ka + 1) / D, ry = (ka + 1) % D;
      const float* px = Xb + (size_t)qx * strideC + (size_t)rx * ncols;
      const float* py = Xb + (size_t)qy * strideC + (size_t)ry * ncols;
      if ((kb & 63) == 0) __builtin_prefetch(px + 64, 0, 1);   // global_prefetch_b8
      b.x = *px;
      b.y = *py;
    }
#pragma unroll
    for (int m = 0; m < MT; ++m) {
      v2f a = *(const v2f*)(Wp + (size_t)m * 16 * K + ka);    // b64, aligned (K even)
      acc[m] = __builtin_amdgcn_wmma_f32_16x16x4_f32(false, a, false, b,
                                                     (short)0, acc[m], false, false);
    }
  }
  const int col = n0 + lm;
#pragma unroll
  for (int m = 0; m < MT; ++m) {
#pragma unroll
    for (int j = 0; j < 8; ++j) {
      const int row = m * 16 + j + hi * 8;   // C/D layout: VGPR j = rows j and j+8
      float v = acc[m][j] + bias[row];
      float* yp = Y + (size_t)row * ldY + col;
      if (mode == 1) v += *yp;
      if (mode == 2) v = fmaxf(v, 0.0f);
      *yp = v;
    }
  }
}

// ---------------------------------------------------------------------------
// Adjacency scores: adj = relu(tanh(alpha*(n1*n2^T - n2*n1^T))), K=40, via WMMA.
// Edge tiles: out-of-range rows/cols are clamped to row 0 (no zero-fill needed:
// A-row m only feeds output row m, B-col n only output col n, and those
// outputs are discarded by the store guard).
// ---------------------------------------------------------------------------
__global__ void adj_score_wmma(const float* __restrict__ n1, const float* __restrict__ n2,
                               float* __restrict__ adj, int N, int Kd, float alpha) {
  const int w0 = blockIdx.x * 16;
  const int v0 = blockIdx.y * 16;
  const int lane = threadIdx.x;
  const int lm = lane & 15;
  const int hi = lane >> 4;
  const int vr = v0 + lm;
  const int wr = w0 + lm;
  const int vc = (vr < N) ? vr : 0;
  const int wc = (wr < N) ? wr : 0;
  v8f c1 = {}, c2 = {};
  for (int kb = 0; kb < Kd; kb += 4) {
    const int ka = kb + hi * 2;
    v2f a1 = *(const v2f*)(n1 + vc * Kd + ka);
    v2f a2 = *(const v2f*)(n2 + vc * Kd + ka);
    v2f b1 = *(const v2f*)(n2 + wc * Kd + ka);
    v2f b2 = *(const v2f*)(n1 + wc * Kd + ka);
    c1 = __builtin_amdgcn_wmma_f32_16x16x4_f32(false, a1, false, b1, (short)0, c1, false, false);
    c2 = __builtin_amdgcn_wmma_f32_16x16x4_f32(false, a2, false, b2, (short)0, c2, false, false);
  }
#pragma unroll
  for (int j = 0; j < 8; ++j) {
    const int row = v0 + j + hi * 8;
    const int col = w0 + lm;
    if (row < N && col < N)
      adj[(size_t)row * N + col] = fmaxf(tanhf(alpha * (c1[j] - c2[j])), 0.0f);
  }
}

// ---------------------------------------------------------------------------
// Top-K threshold mask: one wave32 per row, 20 max-extraction passes in LDS.
// ---------------------------------------------------------------------------
__global__ void topk_thresh_mask(const float* __restrict__ adj, float* __restrict__ madj,
                                 int N, int Ktop) {
  __shared__ float vals[1024];
  const int row = blockIdx.x;
  const int lane = threadIdx.x;
  const float* src = adj + (size_t)row * N;
  for (int i = lane; i < N; i += 32) vals[i] = src[i];
  __syncthreads();
  float thr = 0.0f;
  for (int pass = 0; pass < Ktop; ++pass) {
    float m = -1e30f; int mi = 0;
    for (int i = lane; i < N; i += 32) {
      float v = vals[i];
      if (v > m) { m = v; mi = i; }
    }
    float wm = m;
    for (int off = 16; off > 0; off >>= 1) wm = fmaxf(wm, __shfl_xor(wm, off, 32));
    unsigned long long msk = __ballot(m == wm);
    int leader = __ffsll(msk) - 1;
    if (lane == leader) vals[mi] = -1e30f;   // remove one instance of the max
    __syncthreads();
    thr = wm;                                 // after Ktop passes: Ktop-th largest
  }
  for (int i = lane; i < N; i += 32) {
    float v = src[i];
    madj[(size_t)row * N + i] = (v >= thr) ? v : 0.0f;
  }
}

// CSR (fixed-width 32) of row-normalized (adj [+ transpose] + I)
__global__ void csr_build(const float* __restrict__ madj, int N, int transpose,
                          int* __restrict__ nidx, float* __restrict__ nw, int* __restrict__ nnz) {
  int v = blockIdx.x * blockDim.x + threadIdx.x;
  if (v >= N) return;
  int cnt = 0; float s = 0.0f;
  for (int w = 0; w < N; ++w) {
    float val = transpose ? madj[(size_t)w * N + v] : madj[(size_t)v * N + w];
    if (val != 0.0f && cnt < 31) { nidx[v * 32 + cnt] = w; nw[v * 32 + cnt] = val; s += val; ++cnt; }
  }
  nidx[v * 32 + cnt] = v; nw[v * 32 + cnt] = 1.0f; s += 1.0f; ++cnt;  // self loop
  float inv = 1.0f / s;
  for (int j = 0; j < cnt; ++j) nw[v * 32 + j] *= inv;
  nnz[v] = cnt;
}

// h_out = alpha*X0 + (1-alpha)*(A_norm @ X_in); 4 channels per thread so the
// CSR index/weight loads amortize 4x (memory-bound sparse gather).
__global__ void mixprop_step(const float* __restrict__ X0, const float* __restrict__ Xin,
                             float* __restrict__ Out,
                             const int* __restrict__ nidx, const float* __restrict__ nw,
                             const int* __restrict__ nnz, float alpha) {
  int i = blockIdx.x * blockDim.x + threadIdx.x;   // over (32/4)*cBN
  if (i >= 8 * cBN) return;
  int cg = i / cBN;
  int bn = i - cg * cBN;
  int b = bn / cN;
  int v = bn - b * cN;
  const float* xr = Xin + (size_t)(cg * 4) * cBN + (size_t)b * cN;
  float a0 = 0.f, a1 = 0.f, a2 = 0.f, a3 = 0.f;
  int k = nnz[v];
  for (int j = 0; j < k; ++j) {
    float w = nw[v * 32 + j];
    int id = nidx[v * 32 + j];
    a0 += w * xr[id];
    a1 += w * xr[id + cBN];
    a2 += w * xr[id + 2 * cBN];
    a3 += w * xr[id + 3 * cBN];
  }
  size_t o = (size_t)(cg * 4) * cBN + bn;
  const float ia = 1.0f - alpha;
  Out[o]           = alpha * X0[o]           + ia * a0;
  Out[o + cBN]     = alpha * X0[o + cBN]     + ia * a1;
  Out[o + 2 * cBN] = alpha * X0[o + 2 * cBN] + ia * a2;
  Out[o + 3 * cBN] = alpha * X0[o + 3 * cBN] + ia * a3;
}

// node features: out = tanh(alpha*(emb @ W + b)), 1000x40, K=40
__global__ void node_feat(const float* __restrict__ emb, const float* __restrict__ Wl,
                          const float* __restrict__ bl, float* __restrict__ out) {
  int i = blockIdx.x * blockDim.x + threadIdx.x;
  if (i >= cN * cND) return;
  int row = i / cND, j = i - row * cND;
  float a = bl[j];
  for (int k = 0; k < cND; ++k) a += emb[row * cND + k] * Wl[k * cND + j];
  out[i] = tanhf(cALPHA * a);
}

// combined inception weight pack: Wc[64][32*7]; rows 0..31 filter, 32..63 gate
__global__ void pack_wc(const float* f0, const float* f1, const float* f2, const float* f3,
                        const float* g0, const float* g1, const float* g2, const float* g3,
                        float* __restrict__ Wcl) {
  int i = blockIdx.x * blockDim.x + threadIdx.x;
  if (i >= 64 * 224) return;
  int row = i / 224, k = i - row * 224;
  int ci = k / 7, dt = k - ci * 7;
  int rr = row & 31;
  bool gate = row >= 32;
  int grp = rr >> 3, oc = rr & 7;
  const int ks[4] = {2, 3, 6, 7};
  const float* tab[8] = {f0, f1, f2, f3, g0, g1, g2, g3};
  const float* src = tab[grp + (gate ? 4 : 0)];
  int ksz = ks[grp];
  int j = dt - (7 - ksz);                     // kernel k uses dt in [7-k, 6]
  Wcl[i] = (j >= 0) ? src[(oc * 32 + ci) * ksz + j] : 0.0f;
}

__global__ void pack_bias(const float* b0, const float* b1, const float* b2, const float* b3,
                          const float* c0, const float* c1, const float* c2, const float* c3,
                          float* __restrict__ Wbl) {
  int i = threadIdx.x;
  if (i >= 64) return;
  int rr = i & 31;
  const float* tab[8] = {b0, b1, b2, b3, c0, c1, c2, c3};
  Wbl[i] = tab[(rr >> 3) + (i >= 32 ? 4 : 0)][rr & 7];
}

// pad end2 W [12,128] -> [16,128], bias [12] -> [16] (zeros) so M % 16 == 0
__global__ void pad_end2(const float* __restrict__ w, const float* __restrict__ b,
                         float* __restrict__ wp, float* __restrict__ bp) {
  int i = blockIdx.x * blockDim.x + threadIdx.x;
  if (i < 16 * 128) wp[i] = (i < 12 * 128) ? w[i] : 0.0f;
  if (i < 16) bp[i] = (i < 12) ? b[i] : 0.0f;
}

// x [B,2,N,12] -> xin [2][19][B*N], left zero pad 7
__global__ void build_xin(const float* __restrict__ x, float* __restrict__ xin) {
  int i = blockIdx.x * blockDim.x + threadIdx.x;
  if (i >= cIN * cRF * cBN) return;
  int ci = i / (cRF * cBN);
  int r = i - ci * cRF * cBN;
  int t = r / cBN;
  int bn = r - t * cBN;
  int b = bn / cN, n = bn - b * cN;
  int ts = t - (cRF - cSEQ);
  xin[i] = (ts >= 0) ? x[(((size_t)b * cIN + ci) * cN + n) * cSEQ + ts] : 0.0f;
}

// 1x1 start conv, K=2 (too small for WMMA), float4
__global__ void start_conv(const float* __restrict__ xin, const float* __restrict__ w,
                           const float* __restrict__ b, float* __restrict__ h) {
  int i = blockIdx.x * blockDim.x + threadIdx.x;
  if (i >= cRC * cRF * cBN / 4) return;
  int c = i / (cRF * cBN / 4);
  int r = i - c * (cRF * cBN / 4);
  v4f x0 = *(const v4f*)(xin + (size_t)r * 4);
  v4f x1 = *(const v4f*)(xin + (size_t)cRF * cBN + (size_t)r * 4);
  v4f y = w[c * 2] * x0 + w[c * 2 + 1] * x1 + b[c];
  *(v4f*)(h + ((size_t)c * cRF * cBN / 4 + r) * 4) = y;
}

// skip0 conv: [2,19] window -> [64][BN]  (K=38, not /4, scalar-per-lane float4)
__global__ void skip0_conv(const float* __restrict__ xin, const float* __restrict__ w,
                           const float* __restrict__ b, float* __restrict__ skip) {
  int i = blockIdx.x * blockDim.x + threadIdx.x;
  if (i >= cSC * cBN / 4) return;
  int o = i / (cBN / 4), bn4 = i - o * (cBN / 4);
  v4f acc = {b[o], b[o], b[o], b[o]};
  for (int ci = 0; ci < cIN; ++ci)
    for (int t = 0; t < cRF; ++t) {
      v4f xv = *(const v4f*)(xin + ((size_t)ci * cRF + t) * cBN + (size_t)bn4 * 4);
      acc += w[(o * cIN + ci) * cRF + t] * xv;
    }
  *(v4f*)(skip + (size_t)o * cBN + (size_t)bn4 * 4) = acc;
}

// h = tanh(filter)*sigmoid(gate) from comb[64][BN] -> hg slice (float4)
__global__ void gate_act(const float* __restrict__ comb, float* __restrict__ dst, long dstStride) {
  int i = blockIdx.x * blockDim.x + threadIdx.x;
  if (i >= 32 * cBN / 4) return;
  int c = i / (cBN / 4), x = i - c * (cBN / 4);
  v4f f = *(const v4f*)(comb + (size_t)c * cBN + (size_t)x * 4);
  v4f g = *(const v4f*)(comb + (size_t)(c + 32) * cBN + (size_t)x * 4);
  v4f y;
#pragma unroll
  for (int q = 0; q < 4; ++q) y[q] = tanhf(f[q]) * (1.0f / (1.0f + expf(-g[q])));
  *(v4f*)(dst + (size_t)c * dstStride + (size_t)x * 4) = y;
}

__global__ void copy_slab(float* __restrict__ dst, long dstStride,
                          const float* __restrict__ src, long srcStride, int C) {
  int i = blockIdx.x * blockDim.x + threadIdx.x;
  if (i >= C * cBN / 4) return;
  int c = i / (cBN / 4), x = i - c * (cBN / 4);
  *(v4f*)(dst + (size_t)c * dstStride + (size_t)x * 4) =
      *(const v4f*)(src + (size_t)c * srcStride + (size_t)x * 4);
}

__global__ void residual_add(float* __restrict__ Hn, const float* __restrict__ Hp,
                             int C, int Tout, int Tin) {
  int i = blockIdx.x * blockDim.x + threadIdx.x;
  if (i >= C * Tout * cBN / 4) return;
  int c = i / (Tout * cBN / 4);
  int r = i - c * (Tout * cBN / 4);
  int t = r / (cBN / 4);
  int x = r - t * (cBN / 4);
  v4f* dp = (v4f*)(Hn + ((size_t)c * Tout + t) * cBN + (size_t)x * 4);
  const v4f* sp = (const v4f*)(Hp + ((size_t)c * Tin + t + (Tin - Tout)) * cBN + (size_t)x * 4);
  *dp += *sp;
}

// layernorm stats per batch sample over (C,N,T) — float4 strided loop
__global__ void ln_stats(const float* __restrict__ H, int C, int T,
                         float* __restrict__ mean, float* __restrict__ rstd) {
  const int b = blockIdx.x;
  const int tid = threadIdx.x;
  const int per4 = C * T * (cN / 4);
  float s = 0.f, s2 = 0.f;
  for (int i = tid; i < per4; i += 256) {
    int c = i / (T * (cN / 4));
    int r = i - c * T * (cN / 4);
    int t = r / (cN / 4);
    int n4 = r - t * (cN / 4);
    v4f v = *(const v4f*)(H + ((size_t)c * T + t) * cBN + (size_t)b * cN + (size_t)n4 * 4);
#pragma unroll
    for (int q = 0; q < 4; ++q) { s += v[q]; s2 += v[q] * v[q]; }
  }
  __shared__ float ss[256], sq[256];
  ss[tid] = s; sq[tid] = s2;
  __syncthreads();
  for (int o = 128; o > 0; o >>= 1) {
    if (tid < o) { ss[tid] += ss[tid + o]; sq[tid] += sq[tid + o]; }
    __syncthreads();
  }
  if (tid == 0) {
    float per = (float)(C * T * cN);
    float m = ss[0] / per;
    float var = sq[0] / per - m * m;
    mean[b] = m;
    rstd[b] = rsqrtf(var + 1e-5f);
  }
}

__global__ void ln_apply(float* __restrict__ H, const float* __restrict__ mean,
                         const float* __restrict__ rstd, const float* __restrict__ nw,
                         const float* __restrict__ nb, int C, int T) {
  int i = blockIdx.x * blockDim.x + threadIdx.x;
  if (i >= C * T * cBN / 4) return;
  int c = i / (T * cBN / 4);
  int r = i - c * (T * cBN / 4);
  int t = r / (cBN / 4);
  int x = r - t * (cBN / 4);
  int bn = x * 4;
  int b = bn / cN;
  int n = bn - b * cN;
  float* hp = H + ((size_t)c * T + t) * cBN + bn;
  v4f v = *(const v4f*)hp;
  v4f y;
#pragma unroll
  for (int q = 0; q < 4; ++q) {
    size_t wi = ((size_t)c * cN + n + q) * T + t;   // norm_w shape [RC,N,T]
    y[q] = (v[q] - mean[b]) * rstd[b] * nw[wi] + nb[wi];
  }
  *(v4f*)hp = y;
}

__global__ void relu_ip(float* __restrict__ p, int n4) {
  int i = blockIdx.x * blockDim.x + threadIdx.x;
  if (i < n4) {
    v4f v = *(const v4f*)(p + (size_t)i * 4);
    v4f y = {fmaxf(v[0], 0.f), fmaxf(v[1], 0.f), fmaxf(v[2], 0.f), fmaxf(v[3], 0.f)};
    *(v4f*)(p + (size_t)i * 4) = y;
  }
}

// tmp[16][B*N] -> out [B][OUT=12][N]  (float4 over n; 1000 % 4 == 0)
__global__ void final_out(const float* __restrict__ tmp, float* __restrict__ out) {
  int i = blockIdx.x * blockDim.x + threadIdx.x;
  if (i >= cB * cOUT * (cN / 4)) return;
  int b = i / (cOUT * (cN / 4));
  int r = i - b * cOUT * (cN / 4);
  int o = r / (cN / 4);
  int n4 = r - o * (cN / 4);
  *(v4f*)(out + ((size_t)b * cOUT + o) * cN + (size_t)n4 * 4) =
      *(const v4f*)(tmp + (size_t)o * cBN + (size_t)b * cN + (size_t)n4 * 4);
}

// ---------------------------------------------------------------------------
extern "C" void kernel_launch(void* const* d_in, const int* in_sizes, int n_in,
                              void* d_out, int out_size, void* d_ws, size_t ws_size,
                              hipStream_t stream) {
  (void)in_sizes; (void)n_in; (void)out_size; (void)ws_size;
  auto F = [&](int i) { return (const float*)d_in[i]; };
  // input indices (setup_inputs insertion order):
  // 0 x, 1 idx, 2 emb1, 3 emb2, 4 lin1_w, 5 lin1_b, 6 lin2_w, 7 lin2_b,
  // 8 start_w, 9 start_b, 10 skip0_w, 11 skip0_b,
  // 12..23 filter_w[l][k], 24..35 filter_b, 36..47 gate_w, 48..59 gate_b,
  // 60..62 skip_w, 63..65 skip_b, 66..68 g1_w, 69..71 g1_b, 72..74 g2_w, 75..77 g2_b,
  // 78..80 norm_w, 81..83 norm_b, 84 skipE_w, 85 skipE_b, 86 end1_w, 87 end1_b,
  // 88 end2_w, 89 end2_b

  float* Wsp = (float*)d_ws;
  size_t off = 0;
  auto alloc = [&](size_t nf) { float* p = Wsp + off; off += nf; return p; };
  float* xin  = alloc((size_t)cIN * cRF * cBN);
  float* hA   = alloc((size_t)cRC * cRF * cBN);
  float* hB   = alloc((size_t)cRC * cRF * cBN);
  float* hg   = alloc((size_t)cRC * 13 * cBN);
  float* comb = alloc((size_t)64 * cBN);
  float* cat1 = alloc((size_t)96 * cBN);
  float* cat2 = alloc((size_t)96 * cBN);
  float* skip = alloc((size_t)cSC * cBN);
  float* e1   = alloc((size_t)cEC * cBN);
  float* n1b  = alloc((size_t)cN * cND);
  float* n2b  = alloc((size_t)cN * cND);
  float* adjb = alloc((size_t)cN * cN);
  float* madjb = alloc((size_t)cN * cN);
  int*   idx1 = (int*)alloc(cN * 32);
  float* w1   = alloc(cN * 32);
  int*   nnz1 = (int*)alloc(cN);
  int*   idx2 = (int*)alloc(cN * 32);
  float* w2   = alloc(cN * 32);
  int*   nnz2 = (int*)alloc(cN);
  float* Wc   = alloc(3 * 64 * 224);
  float* Wb   = alloc(3 * 64);
  float* w2p  = alloc(16 * 128);         // padded end2 weights
  float* b2p  = alloc(16);               // padded end2 bias
  float* meanb = alloc(cB);
  float* rstdb = alloc(cB);
  float* tmpo  = alloc((size_t)16 * cBN);

  auto nbl = [](long total, int bs) { return dim3((unsigned)((total + bs - 1) / bs)); };

  // --- adjacency ---
  node_feat<<<nbl((long)cN * cND, 256), 256, 0, stream>>>(F(2), F(4), F(5), n1b);
  node_feat<<<nbl((long)cN * cND, 256), 256, 0, stream>>>(F(3), F(6), F(7), n2b);
  {
    dim3 g((cN + 15) / 16, (cN + 15) / 16);
    adj_score_wmma<<<g, 32, 0, stream>>>(n1b, n2b, adjb, cN, cND, cALPHA);
  }
  topk_thresh_mask<<<cN, 32, 0, stream>>>(adjb, madjb, cN, cK);
  csr_build<<<nbl(cN, 128), 128, 0, stream>>>(madjb, cN, 0, idx1, w1, nnz1);
  csr_build<<<nbl(cN, 128), 128, 0, stream>>>(madjb, cN, 1, idx2, w2, nnz2);

  // --- pack inception weights/biases, padded end2 ---
  for (int l = 0; l < 3; ++l) {
    pack_wc<<<nbl(64 * 224, 256), 256, 0, stream>>>(
        F(12 + l * 4), F(13 + l * 4), F(14 + l * 4), F(15 + l * 4),
        F(36 + l * 4), F(37 + l * 4), F(38 + l * 4), F(39 + l * 4), Wc + l * 64 * 224);
    pack_bias<<<1, 64, 0, stream>>>(
        F(24 + l * 4), F(25 + l * 4), F(26 + l * 4), F(27 + l * 4),
        F(48 + l * 4), F(49 + l * 4), F(50 + l * 4), F(51 + l * 4), Wb + l * 64);
  }
  pad_end2<<<nbl(16 * 128, 256), 256, 0, stream>>>(F(88), F(89), w2p, b2p);

  // --- stem ---
  build_xin<<<nbl((long)cIN * cRF * cBN, 256), 256, 0, stream>>>(F(0), xin);
  start_conv<<<nbl((long)cRC * cRF * cBN / 4, 256), 256, 0, stream>>>(xin, F(8), F(9), hA);
  skip0_conv<<<nbl((long)cSC * cBN / 4, 256), 256, 0, stream>>>(xin, F(10), F(11), skip);

  // GEMM dispatch: MT = M/16 and D are compile-time specializations
  auto gemm = [&](const float* Wm, const float* X, float* Y, const float* bias,
                  int M, int K, int D, int strideC, int t0, int ldY, int mode) {
    dim3 g(cBN / 64), t(32, 4);
    const int key = (M / 16) * 100 + D;
    switch (key) {
      case 407: wmma_gemm_f32<4, 7><<<g, t, 0, stream>>>(Wm, X, Y, bias, K, strideC, t0, cBN, ldY, mode); break;
      case 413: wmma_gemm_f32<4, 13><<<g, t, 0, stream>>>(Wm, X, Y, bias, K, strideC, t0, cBN, ldY, mode); break;
      case 401: wmma_gemm_f32<4, 1><<<g, t, 0, stream>>>(Wm, X, Y, bias, K, strideC, t0, cBN, ldY, mode); break;
      case 201: wmma_gemm_f32<2, 1><<<g, t, 0, stream>>>(Wm, X, Y, bias, K, strideC, t0, cBN, ldY, mode); break;
      case 801: wmma_gemm_f32<8, 1><<<g, t, 0, stream>>>(Wm, X, Y, bias, K, strideC, t0, cBN, ldY, mode); break;
      case 101: wmma_gemm_f32<1, 1><<<g, t, 0, stream>>>(Wm, X, Y, bias, K, strideC, t0, cBN, ldY, mode); break;
      default: break;
    }
  };

  const int TIN[3] = {19, 13, 7};
  const int TOUT[3] = {13, 7, 1};
  float* hin = hA;
  float* hout = hB;
  for (int l = 0; l < 3; ++l) {
    const int Tin = TIN[l], Tout = TOUT[l];
    // dilated inception (filter+gate fused, K = 32*7 = 224) + gated activation
    for (int t = 0; t < Tout; ++t) {
      gemm(Wc + l * 64 * 224, hin, comb, Wb + l * 64, 64, 224, 7, Tin * cBN, t, cBN, 0);
      gate_act<<<nbl((long)32 * cBN / 4, 256), 256, 0, stream>>>(comb, hg + (size_t)t * cBN,
                                                                 (long)Tout * cBN);
    }
    // skip accumulation: K = 32*Tout, D = Tout
    gemm(F(60 + l), hg, skip, F(63 + l), cSC, 32 * Tout, Tout, Tout * cBN, 0, cBN, 1);
    // mixprop (both directions) per timestep
    for (int t = 0; t < Tout; ++t) {
      copy_slab<<<nbl((long)32 * cBN / 4, 256), 256, 0, stream>>>(cat1, (long)cBN,
          hg + (size_t)t * cBN, (long)Tout * cBN, 32);
      copy_slab<<<nbl((long)32 * cBN / 4, 256), 256, 0, stream>>>(cat2, (long)cBN,
          hg + (size_t)t * cBN, (long)Tout * cBN, 32);
      mixprop_step<<<nbl((long)8 * cBN, 256), 256, 0, stream>>>(
          cat1, cat1, cat1 + (size_t)32 * cBN, idx1, w1, nnz1, cPROP);
      mixprop_step<<<nbl((long)8 * cBN, 256), 256, 0, stream>>>(
          cat1, cat1 + (size_t)32 * cBN, cat1 + (size_t)64 * cBN, idx1, w1, nnz1, cPROP);
      mixprop_step<<<nbl((long)8 * cBN, 256), 256, 0, stream>>>(
          cat2, cat2, cat2 + (size_t)32 * cBN, idx2, w2, nnz2, cPROP);
      mixprop_step<<<nbl((long)8 * cBN, 256), 256, 0, stream>>>(
          cat2, cat2 + (size_t)32 * cBN, cat2 + (size_t)64 * cBN, idx2, w2, nnz2, cPROP);
      gemm(F(66 + l), cat1, hout + (size_t)t * cBN, F(69 + l), cRC, 96, 1, cBN, 0, Tout * cBN, 0);
      gemm(F(72 + l), cat2, hout + (size_t)t * cBN, F(75 + l), cRC, 96, 1, cBN, 0, Tout * cBN, 1);
    }
    residual_add<<<nbl((long)cRC * Tout * cBN / 4, 256), 256, 0, stream>>>(hout, hin, cRC, Tout, Tin);
    ln_stats<<<cB, 256, 0, stream>>>(hout, cRC, Tout, meanb, rstdb);
    ln_apply<<<nbl((long)cRC * Tout * cBN / 4, 256), 256, 0, stream>>>(hout, meanb, rstdb,
                                                                       F(78 + l), F(81 + l), cRC, Tout);
    float* tp = hin; hin = hout; hout = tp;
  }

  // --- head ---
  gemm(F(84), hin, skip, F(85), cSC, cRC, 1, cBN, 0, cBN, 1);          // skipE += W@h
  relu_ip<<<nbl((long)cSC * cBN / 4, 256), 256, 0, stream>>>(skip, cSC * cBN / 4);
  gemm(F(86), skip, e1, F(87), cEC, cSC, 1, cBN, 0, cBN, 2);           // end1 + relu
  gemm(w2p, e1, tmpo, b2p, 16, cEC, 1, cBN, 0, cBN, 0);                // end2 (padded M=16)
  final_out<<<nbl((long)cB * cOUT * (cN / 4), 256), 256, 0, stream>>>(tmpo, (float*)d_out);
}